// CayleySolitonPropagator_4939212390676
// MI455X (gfx1250) — compile-verified
//
#include <hip/hip_runtime.h>
#include <math.h>

typedef float v2f __attribute__((ext_vector_type(2)));
typedef float v8f __attribute__((ext_vector_type(8)));

#define D_DIM     1024
#define NBLK      64            // D_DIM / 16 block rows
#define NSYS      16            // systems per workgroup (WMMA N dimension)
#define NTHREADS  512
#define NWAVES    16            // 512 / wave32
#define TASKS_PW  4             // 64 block rows / 16 waves
#define HALF_DT   0.05f
#define CG_ITERS  20
#define CG_TOL2   1e-12f

// LDS layout (float offsets)
#define OFF_PR    0
#define OFF_PI    16384
#define OFF_APR   32768
#define OFF_API   49152
#define OFF_HA    65536          // 3 * 16 * 16 tiles (d = -1,0,+1), layout [d][K][M] = A[M][K]
#define OFF_ALPH  66304          // 1024
#define OFF_DOT   67328          // 16
#define OFF_RS    67344          // 16
#define OFF_TH    67360          // 16
#define OFF_ACG   67376          // 16
#define OFF_BET   67392          // 16
#define OFF_ACT   67408          // 16
#define LDS_FLOATS 67424

// out_r = v_r + sign*h*(H v_i) ; out_i = v_i - sign*h*(H v_r)
// sign=+1 : rhs = (I - i*dt/2*H) v   ; sign=-1 : a_minus = (I + i*dt/2*H) v
// Tridiagonal 16x16 tiles (d=-1,0,+1) go through the fp32 matrix pipe; the
// 4-entry corners of the offset-20 band that spill into blocks d=+-2 are
// applied as scalar FMAs straight into the D-fragment accumulators.
// When do_pap is set, each lane also accumulates its column's partial of
// dot(p, Ap) from the in-register fragments and adds it to sDot once.
__device__ __forceinline__ void ham_apply(float* sm, const v2f hA[3][4],
                                          float w20, float sign, bool do_pap,
                                          int wave, int lane) {
  const float h = sign * HALF_DT;
  const int m     = lane & 15;
  const bool lo   = (lane < 16);
  const int khalf = lo ? 0 : 2;
  float* Pr  = sm + OFF_PR;  float* Pi  = sm + OFF_PI;
  float* APr = sm + OFF_APR; float* APi = sm + OFF_API;
  float pap = 0.f;
  #pragma unroll 1
  for (int q = 0; q < TASKS_PW; ++q) {
    const int i = q * NWAVES + wave;            // block row 0..63
    v8f accR = {}; v8f accI = {};
    #pragma unroll
    for (int d = 0; d < 3; ++d) {
      const int cb = (i + d - 1 + NBLK) & (NBLK - 1);   // circulant column block i-1..i+1
      const int rowbase = cb * 16;
      #pragma unroll
      for (int j = 0; j < 4; ++j) {
        const int K    = rowbase + 4 * j + khalf;
        const int bidx = K * NSYS + m;
        v2f bR, bI;
        bR.x = Pr[bidx]; bR.y = Pr[bidx + NSYS];
        bI.x = Pi[bidx]; bI.y = Pi[bidx + NSYS];
        accR = __builtin_amdgcn_wmma_f32_16x16x4_f32(false, hA[d][j], false, bR,
                                                     (short)0, accR, false, false);
        accI = __builtin_amdgcn_wmma_f32_16x16x4_f32(false, hA[d][j], false, bI,
                                                     (short)0, accI, false, false);
      }
    }
    // offset-20 corner corrections from blocks i-2 / i+2 (4 entries each, value -w20).
    // lanes 0-15 own M=k (k=0..3 -> rows cbm*16+12+q2); lanes 16-31 own M=k+8
    // (k=4..7 -> rows cbp*16+q2). All WMMAs for this task are already issued.
    {
      const int cbm = (i - 2 + NBLK) & (NBLK - 1);
      const int cbp = (i + 2) & (NBLK - 1);
      const int crow = lo ? (cbm * 16 + 12) : (cbp * 16);
      float cr[4], ci[4];
      #pragma unroll
      for (int q2 = 0; q2 < 4; ++q2) {
        const int idx = (crow + q2) * NSYS + m;
        cr[q2] = Pr[idx];
        ci[q2] = Pi[idx];
      }
      if (lo) {
        accR[0] -= w20 * cr[0]; accR[1] -= w20 * cr[1];
        accR[2] -= w20 * cr[2]; accR[3] -= w20 * cr[3];
        accI[0] -= w20 * ci[0]; accI[1] -= w20 * ci[1];
        accI[2] -= w20 * ci[2]; accI[3] -= w20 * ci[3];
      } else {
        accR[4] -= w20 * cr[0]; accR[5] -= w20 * cr[1];
        accR[6] -= w20 * cr[2]; accR[7] -= w20 * cr[3];
        accI[4] -= w20 * ci[0]; accI[5] -= w20 * ci[1];
        accI[6] -= w20 * ci[2]; accI[7] -= w20 * ci[3];
      }
    }
    // combine with identity term, store Ap, and (optionally) accumulate p.Ap
    // (C/D layout: vgpr k -> M=k(+8), lane%16 -> N)
    #pragma unroll
    for (int k = 0; k < 8; ++k) {
      const int M   = k + (lo ? 0 : 8);
      const int idx = (i * 16 + M) * NSYS + m;
      const float vr = Pr[idx], vi = Pi[idx];
      const float apr = vr + h * accI[k];
      const float api = vi - h * accR[k];
      APr[idx] = apr;
      APi[idx] = api;
      if (do_pap) pap += vr * apr + vi * api;
    }
  }
  if (do_pap) atomicAdd(&sm[OFF_DOT + m], pap);
}

__global__ __launch_bounds__(NTHREADS, 1)
void cayley_cg_kernel(const float* __restrict__ psi_r, const float* __restrict__ psi_i,
                      const float* __restrict__ alpha_v, const float* __restrict__ ham_w,
                      float* __restrict__ out) {
  extern __shared__ float sm[];
  const int t    = threadIdx.x;
  const int wave = t >> 5, lane = t & 31;
  const int n    = t & 15;          // system column owned in vector phases
  const int g    = t >> 4;          // 0..31 row-group
  const int wg   = blockIdx.x;

  float* Pr  = sm + OFF_PR;  float* Pi  = sm + OFF_PI;
  float* APr = sm + OFF_APR; float* APi = sm + OFF_API;
  float* HA  = sm + OFF_HA;  float* sAl = sm + OFF_ALPH;
  float* sDot = sm + OFF_DOT; float* sRs = sm + OFF_RS;  float* sTh  = sm + OFF_TH;
  float* sAcg = sm + OFF_ACG; float* sBet = sm + OFF_BET; float* sAct = sm + OFF_ACT;

  // ---- build the 3 tridiagonal circulant 16x16 H tiles (HA[d][K][M] = A[M][K]) ----
  float sumW = 0.f;
  #pragma unroll
  for (int e = 0; e < 15; ++e) sumW += ham_w[e];
  const float w20 = ham_w[14];       // offset 20 = 2^2 * (4+1): only (m=2,j=4)
  for (int e = t; e < 768; e += NTHREADS) {
    const int d3 = e >> 8, rem = e & 255, K = rem >> 4, M = rem & 15;
    const int delta = (d3 - 1) * 16 + K - M;   // |delta| <= 31: no circulant aliasing
    float v;
    if (delta == 0) {
      v = 2.0f * sumW;
    } else {
      const int ad = delta < 0 ? -delta : delta;
      v = 0.f;
      #pragma unroll
      for (int mm = 0; mm < 3; ++mm)
        #pragma unroll
        for (int jj = 0; jj < 5; ++jj)
          if ((1 << mm) * (jj + 1) == ad) v -= ham_w[mm * 5 + jj];
    }
    HA[e] = v;
  }
  if (t < 16) sDot[t] = 0.f;
  for (int e = t; e < D_DIM; e += NTHREADS) sAl[e] = alpha_v[e];

  // ---- load this WG's 16 contiguous systems (coalesced) into LDS [row][sys] ----
  const float* gr = psi_r + (size_t)wg * (NSYS * D_DIM);
  const float* gi = psi_i + (size_t)wg * (NSYS * D_DIM);
  #pragma unroll
  for (int k = 0; k < 32; ++k) {
    const int idx = t + k * NTHREADS;              // 0..16383
    const int sys = idx >> 10, row = idx & 1023;
    const int l = row * NSYS + sys;
    Pr[l] = gr[idx];
    Pi[l] = gi[idx];
  }
  __syncthreads();

  // ---- preload per-lane A-fragments for the 3 tiles (16x4 f32 layout) ----
  v2f hA[3][4];
  {
    const int m = lane & 15;
    const int khalf = (lane < 16) ? 0 : 2;
    #pragma unroll
    for (int d = 0; d < 3; ++d)
      #pragma unroll
      for (int j = 0; j < 4; ++j) {
        const int K0 = 4 * j + khalf;
        hA[d][j].x = HA[(d * 16 + K0) * 16 + m];
        hA[d][j].y = HA[(d * 16 + K0 + 1) * 16 + m];
      }
  }

  // ---- intensity mean per system ----
  {
    float acc = 0.f;
    #pragma unroll
    for (int k = 0; k < 32; ++k) {
      const int idx = (g * 32 + k) * NSYS + n;
      const float a = Pr[idx], b = Pi[idx];
      acc += a * a + b * b;
    }
    atomicAdd(&sDot[n], acc);
  }
  __syncthreads();
  if (t < 16) sRs[t] = 1.0f / (sDot[t] * (1.0f / (float)D_DIM) + 1e-8f);  // temp: 1/(mean+eps)
  __syncthreads();

  // ---- nonlinear phase rotation in place ----
  {
    const float inv = sRs[n];
    #pragma unroll
    for (int k = 0; k < 32; ++k) {
      const int row = g * 32 + k;
      const int idx = row * NSYS + n;
      const float a = Pr[idx], b = Pi[idx];
      const float ph = sAl[row] * ((a * a + b * b) * inv);
      float sv, cv;
      sincosf(ph, &sv, &cv);
      Pr[idx] = a * cv - b * sv;
      Pi[idx] = a * sv + b * cv;
    }
  }
  if (t < 16) sDot[t] = 0.f;
  __syncthreads();

  // ---- rhs = (I - i*dt/2*H) rot  -> APr/APi ----
  ham_apply(sm, hA, w20, +1.0f, false, wave, lane);
  __syncthreads();

  // ---- CG init: x=0, r=b, p=r, rs=r.r ----
  float xr[32], xi[32], rr[32], ri[32];
  {
    float acc = 0.f;
    #pragma unroll
    for (int k = 0; k < 32; ++k) {
      const int idx = (g * 32 + k) * NSYS + n;
      const float br = APr[idx], bi = APi[idx];
      rr[k] = br; ri[k] = bi; xr[k] = 0.f; xi[k] = 0.f;
      acc += br * br + bi * bi;
      Pr[idx] = br; Pi[idx] = bi;          // p = r
    }
    atomicAdd(&sDot[n], acc);
  }
  __syncthreads();
  if (t < 16) {
    const float rs0 = sDot[t];
    sRs[t] = rs0;
    sTh[t] = CG_TOL2 * fmaxf(rs0, 1e-30f);
    sDot[t] = 0.f;                   // ready for iteration 0's fused p.Ap
  }
  __syncthreads();

  // ---- 20 fixed CG iterations with per-system convergence masking ----
  for (int iter = 0; iter < CG_ITERS; ++iter) {
    // Ap = (I + i*dt/2*H) p, with fused per-system dot(p, Ap) into sDot
    ham_apply(sm, hA, w20, -1.0f, true, wave, lane);
    __syncthreads();
    if (t < 16) {
      const float pap = sDot[t];
      const float act = (sRs[t] > sTh[t]) ? 1.0f : 0.0f;
      sAct[t] = act;
      sAcg[t] = act * (sRs[t] / (pap + 1e-30f));
      sDot[t] = 0.f;                 // ready for rs_new accumulation
    }
    __syncthreads();
    // x += a*p ; r -= a*Ap ; rs_new = r.r
    {
      const float a = sAcg[n];
      float acc = 0.f;
      #pragma unroll
      for (int k = 0; k < 32; ++k) {
        const int idx = (g * 32 + k) * NSYS + n;
        xr[k] += a * Pr[idx];
        xi[k] += a * Pi[idx];
        rr[k] -= a * APr[idx];
        ri[k] -= a * APi[idx];
        acc += rr[k] * rr[k] + ri[k] * ri[k];
      }
      atomicAdd(&sDot[n], acc);
    }
    __syncthreads();
    if (t < 16) {
      const float rsn = sDot[t];
      if (sAct[t] > 0.5f) { sBet[t] = rsn / (sRs[t] + 1e-30f); sRs[t] = rsn; }
      else                { sBet[t] = 0.0f; }
      sDot[t] = 0.f;                 // ready for next iteration's fused p.Ap
    }
    __syncthreads();
    // p = active ? r + beta*p : p
    if (sAct[n] > 0.5f) {
      const float b = sBet[n];
      #pragma unroll
      for (int k = 0; k < 32; ++k) {
        const int idx = (g * 32 + k) * NSYS + n;
        Pr[idx] = rr[k] + b * Pr[idx];
        Pi[idx] = ri[k] + b * Pi[idx];
      }
    }
    __syncthreads();
  }

  // ---- writeback x as [B,S,D,2] (float2 per element, per-thread contiguous 256B runs) ----
  {
    float2* o = (float2*)out;
    const size_t sysg = (size_t)wg * NSYS + n;
    #pragma unroll
    for (int k = 0; k < 32; ++k) {
      const int row = g * 32 + k;
      o[sysg * D_DIM + row] = make_float2(xr[k], xi[k]);
    }
  }
}

extern "C" void kernel_launch(void* const* d_in, const int* in_sizes, int n_in,
                              void* d_out, int out_size, void* d_ws, size_t ws_size,
                              hipStream_t stream) {
  const float* psi_r = (const float*)d_in[0];
  const float* psi_i = (const float*)d_in[1];
  const float* alpha = (const float*)d_in[2];
  const float* ham_w = (const float*)d_in[3];
  float* out = (float*)d_out;
  const int nsys   = in_sizes[0] / D_DIM;     // 16384
  const int blocks = nsys / NSYS;             // 1024
  const size_t lds_bytes = (size_t)LDS_FLOATS * sizeof(float);  // ~263 KB of the 320 KB WGP LDS
  cayley_cg_kernel<<<blocks, NTHREADS, lds_bytes, stream>>>(psi_r, psi_i, alpha, ham_w, out);
}